// GraphSAGEConv_73074573574435
// MI455X (gfx1250) — compile-verified
//
#include <hip/hip_runtime.h>

typedef __attribute__((ext_vector_type(2))) float v2f;
typedef __attribute__((ext_vector_type(8))) float v8f;

#define N_NODES 50000
#define D 64

// ---------------------------------------------------------------------------
// Stage 1: edge scatter. 16 threads per edge, 4 contiguous floats per thread.
// agg[row] += x[col] (atomic f32, resolves in L2 since agg is 12.8MB << 192MB)
// deg[row] += 1 (one thread per edge)
// ---------------------------------------------------------------------------
__global__ void sage_scatter(const float* __restrict__ x,
                             const long long* __restrict__ ei,
                             float* __restrict__ agg,
                             float* __restrict__ deg,
                             int E) {
  long long gid = (long long)blockIdx.x * blockDim.x + threadIdx.x;
  int e = (int)(gid >> 4);
  int chunk = (int)(gid & 15);
  if (e >= E) return;
  int row = (int)ei[e];        // destination node
  int col = (int)ei[E + e];    // source node
  const float4 v = *(const float4*)(x + (size_t)col * D + chunk * 4);
  float* dst = agg + (size_t)row * D + chunk * 4;
  atomicAdd(dst + 0, v.x);
  atomicAdd(dst + 1, v.y);
  atomicAdd(dst + 2, v.z);
  atomicAdd(dst + 3, v.w);
  if (chunk == 0) atomicAdd(deg + row, 1.0f);
}

// ---------------------------------------------------------------------------
// Stage 2: out = x @ Wself^T + b_self + (agg/deg) @ Wneigh^T + b_neigh
// One wave32 per 16-row tile; fp32 WMMA 16x16x4 keeps full precision.
// A layout (16x4 f32): lane&15 = M; lanes 0-15 hold K={0,1}, lanes 16-31 K={2,3}.
// B layout (4x16 f32): lane&15 = N; same K-split across lane halves.
// C/D layout (16x16 f32): VGPR g -> rows {g, g+8} selected by lane half; N=lane&15.
// ---------------------------------------------------------------------------
__global__ void __launch_bounds__(256)
sage_gemm(const float* __restrict__ x, const float* __restrict__ agg,
          const float* __restrict__ deg,
          const float* __restrict__ Wself, const float* __restrict__ bself,
          const float* __restrict__ Wneigh, const float* __restrict__ bneigh,
          float* __restrict__ out) {
  const int lane = threadIdx.x & 31;
  const int wave = threadIdx.x >> 5;
  const int tile = blockIdx.x * 8 + wave;          // 16 nodes per wave
  if (tile * 16 >= N_NODES) return;                // wave-uniform exit (EXEC all-1s inside)

  const int m    = lane & 15;                      // row within tile / col within W-tile
  const int kb   = (lane >> 4) * 2;                // K sub-offset for this lane half
  const int node = tile * 16 + m;

  const float dinv = 1.0f / fmaxf(deg[node], 1.0f);

  // Preload A-fragments for both matmul inputs (16 K-steps of 4).
  v2f ax[16], an[16];
#pragma unroll
  for (int kt = 0; kt < 16; ++kt) {
    ax[kt] = *(const v2f*)(x + (size_t)node * D + kt * 4 + kb);
    v2f t  = *(const v2f*)(agg + (size_t)node * D + kt * 4 + kb);
    an[kt].x = t.x * dinv;
    an[kt].y = t.y * dinv;
  }

#pragma unroll
  for (int nt = 0; nt < 4; ++nt) {                 // 4 tiles of 16 output columns
    const int n = nt * 16 + (lane & 15);
    v8f c = {};
#pragma unroll
    for (int kt = 0; kt < 16; ++kt) {              // x @ Wself^T : B[k][n] = Wself[n][k]
      v2f b = *(const v2f*)(Wself + (size_t)n * D + kt * 4 + kb);
      c = __builtin_amdgcn_wmma_f32_16x16x4_f32(false, ax[kt], false, b,
                                                (short)0, c, false, false);
    }
#pragma unroll
    for (int kt = 0; kt < 16; ++kt) {              // agg @ Wneigh^T
      v2f b = *(const v2f*)(Wneigh + (size_t)n * D + kt * 4 + kb);
      c = __builtin_amdgcn_wmma_f32_16x16x4_f32(false, an[kt], false, b,
                                                (short)0, c, false, false);
    }
    const float bias = bself[n] + bneigh[n];
#pragma unroll
    for (int g = 0; g < 8; ++g) {
      const int r = tile * 16 + g + ((lane >> 4) << 3);
      out[(size_t)r * D + n] = c[g] + bias;
    }
  }
}

extern "C" void kernel_launch(void* const* d_in, const int* in_sizes, int n_in,
                              void* d_out, int out_size, void* d_ws, size_t ws_size,
                              hipStream_t stream) {
  const float*     x      = (const float*)d_in[0];
  const long long* ei     = (const long long*)d_in[1];   // int64 [2, E]
  const float*     Wself  = (const float*)d_in[2];
  const float*     bself  = (const float*)d_in[3];
  const float*     Wneigh = (const float*)d_in[4];
  const float*     bneigh = (const float*)d_in[5];
  float*           out    = (float*)d_out;

  const int E = in_sizes[1] / 2;

  float* agg = (float*)d_ws;                       // [N, 64]
  float* deg = agg + (size_t)N_NODES * D;          // [N]

  hipMemsetAsync(d_ws, 0, (size_t)N_NODES * (D + 1) * sizeof(float), stream);

  {
    long long threads = (long long)E * 16;
    int blocks = (int)((threads + 255) / 256);
    sage_scatter<<<blocks, 256, 0, stream>>>(x, ei, agg, deg, E);
  }
  {
    int tiles  = (N_NODES + 15) / 16;
    int blocks = (tiles + 7) / 8;                  // 8 waves (16-row tiles) per block
    sage_gemm<<<blocks, 256, 0, stream>>>(x, agg, deg, Wself, bself,
                                          Wneigh, bneigh, out);
  }
}